// GRUFrame_18872086298653
// MI455X (gfx1250) — compile-verified
//
#include <hip/hip_runtime.h>
#include <hip/hip_bf16.h>
#include <stdint.h>

// ---------------------------------------------------------------------------
// GRUFrame: 2-layer bidirectional GRU (T=1024, B=32, D=256, H=256), fp32.
//   1) gi GEMMs ((T*B) x in_dim x 3H) via native fp32 WMMA 16x16x4 tiles,
//      per-example time roll folded into A-row addressing; K-loop split per
//      source half (no per-iteration address selects).
//   2) Sequential scan kernel (1024 thr = 32 waves per direction):
//      - gi[t] slice (96KB) staged global->LDS with async-LDS loads,
//        overlapped with the recurrent WMMA matmul (h in LDS, Whh from L2)
//      - gh staged in LDS, pointwise gates update h in LDS.
//   3) Finalize: mask by lengths, roll backward outputs, gather h_n.
// ---------------------------------------------------------------------------

typedef float v2f __attribute__((ext_vector_type(2)));
typedef float v8f __attribute__((ext_vector_type(8)));

#define TT 1024
#define BB 32
#define HH 256
#define GG 768           // 3*H
#define H_STRIDE 260     // LDS padding to spread banks (256 % 64 == 0 otherwise)
#define G_STRIDE 772     // LDS padding for gh buffer

__device__ __forceinline__ v8f wmma4(v2f a, v2f b, v8f c) {
  // D = A(16x4,f32) * B(4x16,f32) + C(16x16,f32)
  return __builtin_amdgcn_wmma_f32_16x16x4_f32(
      /*neg_a=*/false, a, /*neg_b=*/false, b,
      /*c_mod=*/(short)0, c, /*reuse_a=*/false, /*reuse_b=*/false);
}

__device__ __forceinline__ int roll_t(int t, int mode, int shift) {
  return (t + mode * shift) & (TT - 1);   // two's-complement AND handles negatives
}

// ---------------------------------------------------------------------------
// gi GEMM: gi[(t,b), g] = sum_k A[(t,b), k] * w[g, k] + bias[g]
// A row = concat(srcF[rollF(t), b, :C0], srcB[rollB(t), b, :C1])
// Tiles: M = T*B = 32768 (2048 tiles), N = 768 (48 tiles). 1 wave per tile.
// ---------------------------------------------------------------------------
__global__ __launch_bounds__(256) void gi_gemm_kernel(
    const float* __restrict__ srcF, const float* __restrict__ srcB,
    int C0, int C1, int modeF, int modeB,
    const float* __restrict__ w, const float* __restrict__ bias,
    const int* __restrict__ lengths, float* __restrict__ gi) {
  const int in_dim = C0 + C1;
  const int lane  = threadIdx.x & 31;
  const int wave  = threadIdx.x >> 5;
  const int tile  = blockIdx.x * 8 + wave;       // 12288 blocks * 8 waves = 98304 tiles
  const int ntile = tile % 48;
  const int mtile = tile / 48;
  const int lidx  = lane & 15;
  const int lhalf = lane >> 4;

  const int m = mtile * 16 + lidx;               // global A row for this lane
  const int t = m >> 5;
  const int b = m & 31;
  const int shift = TT - lengths[b];
  const int tF = roll_t(t, modeF, shift);
  const int tB = roll_t(t, modeB, shift);
  const float* aF = srcF + ((size_t)tF * BB + b) * (size_t)C0 + 2 * lhalf;
  const float* aB = (C1 > 0) ? (srcB + ((size_t)tB * BB + b) * (size_t)C1 + 2 * lhalf)
                             : srcF;

  const int n = ntile * 16 + lidx;               // B column for this lane
  const float* wrow = w + (size_t)n * (size_t)in_dim + 2 * lhalf;

  v8f acc = {0.f, 0.f, 0.f, 0.f, 0.f, 0.f, 0.f, 0.f};
  // First half: srcF (k in [0, C0))  -- no per-iteration source select
  for (int k0 = 0; k0 < C0; k0 += 4) {
    v2f a  = *(const v2f*)(aF + k0);
    v2f bv = *(const v2f*)(wrow + k0);
    acc = wmma4(a, bv, acc);
  }
  // Second half: srcB (k in [C0, in_dim))
  for (int k0 = C0; k0 < in_dim; k0 += 4) {
    v2f a  = *(const v2f*)(aB + (k0 - C0));
    v2f bv = *(const v2f*)(wrow + k0);
    acc = wmma4(a, bv, acc);
  }

  const float bi = bias[n];
  const int colbase = ntile * 16 + lidx;
#pragma unroll
  for (int i = 0; i < 8; ++i) {
    const int row = mtile * 16 + i + 8 * lhalf;  // C/D layout
    gi[(size_t)row * GG + colbase] = acc[i] + bi;
  }
}

// ---------------------------------------------------------------------------
// Recurrent scan: one workgroup (1024 thr = 32 waves) per direction.
// blockIdx.x == 0 -> forward, == 1 -> backward (reverse-time scan).
// Per step: async-stage gi[t] into LDS while computing 96 gh WMMA tiles
// (2 M-tiles x 48 N-tiles; each wave owns 3 N-tiles).
// ---------------------------------------------------------------------------
__global__ __launch_bounds__(1024) void gru_scan_kernel(
    const float* __restrict__ giF, const float* __restrict__ giB,
    const float* __restrict__ whhF, const float* __restrict__ whhB,
    const float* __restrict__ bhhF, const float* __restrict__ bhhB,
    float* __restrict__ ysF, float* __restrict__ ysB) {
  extern __shared__ float lds[];
  float* hS  = lds;                                  // 32 x H_STRIDE
  float* gS  = lds + 32 * H_STRIDE;                  // 32 x G_STRIDE
  float* giS = lds + 32 * H_STRIDE + 32 * G_STRIDE;  // 32 x 768 (raw copy of gi[t])

  const int dir = blockIdx.x;
  const float* gi  = dir ? giB  : giF;
  const float* whh = dir ? whhB : whhF;
  const float* bhh = dir ? bhhB : bhhF;
  float*       ys  = dir ? ysB  : ysF;

  const int tid = threadIdx.x;
  for (int i = tid; i < 32 * H_STRIDE; i += 1024) hS[i] = 0.f;
  __syncthreads();

  const int lane  = tid & 31;
  const int wave  = tid >> 5;
  const int lidx  = lane & 15;
  const int lhalf = lane >> 4;
  const int mt    = wave & 1;          // which 16-row block of the 32-batch
  const int nb    = (wave >> 1) * 3;   // first of 3 N-tiles (48 total)

  const float* hrow = hS + (mt * 16 + lidx) * H_STRIDE;
  const float* w0 = whh + (size_t)((nb + 0) * 16 + lidx) * HH + 2 * lhalf;
  const float* w1 = whh + (size_t)((nb + 1) * 16 + lidx) * HH + 2 * lhalf;
  const float* w2 = whh + (size_t)((nb + 2) * 16 + lidx) * HH + 2 * lhalf;

  // Generic pointers into LDS carry the LDS byte offset in their low 32 bits
  // (flat->LDS truncation, ISA 10.2), usable directly as async-LDS VDST.
  const unsigned giS_lds = (unsigned)(uintptr_t)giS;

  for (int s = 0; s < TT; ++s) {
    const int t = dir ? (TT - 1 - s) : s;
    const float* git = gi + (size_t)t * BB * GG;

    // Kick off async global->LDS staging of gi[t] (6144 x 16B chunks,
    // 6 per thread); overlaps with the recurrent WMMA matmul below.
#pragma unroll
    for (int u = 0; u < 6; ++u) {
      const int chunk = tid + u * 1024;
      const unsigned           la = giS_lds + chunk * 16;
      const unsigned long long ga =
          (unsigned long long)(uintptr_t)git + (unsigned long long)chunk * 16;
      asm volatile("global_load_async_to_lds_b128 %0, %1, off"
                   :: "v"(la), "v"(ga) : "memory");
    }

    v8f a0 = {0.f,0.f,0.f,0.f,0.f,0.f,0.f,0.f};
    v8f a1 = {0.f,0.f,0.f,0.f,0.f,0.f,0.f,0.f};
    v8f a2 = {0.f,0.f,0.f,0.f,0.f,0.f,0.f,0.f};
    for (int k0 = 0; k0 < HH; k0 += 4) {
      const int k = k0 + 2 * lhalf;
      v2f a = *(const v2f*)(hrow + k);         // ds_load_b64, A reused x3
      a0 = wmma4(a, *(const v2f*)(w0 + k0), a0);
      a1 = wmma4(a, *(const v2f*)(w1 + k0), a1);
      a2 = wmma4(a, *(const v2f*)(w2 + k0), a2);
    }

    // Stage gh into LDS for the gate phase.
#pragma unroll
    for (int i = 0; i < 8; ++i) {
      const int row = mt * 16 + i + 8 * lhalf;
      float* gr = gS + row * G_STRIDE;
      gr[(nb + 0) * 16 + lidx] = a0[i];
      gr[(nb + 1) * 16 + lidx] = a1[i];
      gr[(nb + 2) * 16 + lidx] = a2[i];
    }
    // Each wave waits for its own async chunks; the barrier then publishes
    // every wave's LDS writes (async-LDS completion is per-wave ASYNCcnt).
    asm volatile("s_wait_asynccnt 0x0" ::: "memory");
    __syncthreads();

    // Gates: 32*256 elements, 8 per thread. r,z,n gate order (torch GRUCell).
    float* yst = ys + (size_t)t * BB * HH;
#pragma unroll
    for (int u = 0; u < 8; ++u) {
      const int e = tid + u * 1024;
      const int b = e >> 8;
      const int j = e & 255;
      const float* gib = giS + b * GG;          // staged gi[t] in LDS
      const float* ghb = gS + b * G_STRIDE;
      const float hr = ghb[j]       + bhh[j];
      const float hz = ghb[256 + j] + bhh[256 + j];
      const float hn = ghb[512 + j] + bhh[512 + j];
      const float r  = 1.f / (1.f + __expf(-(gib[j] + hr)));
      const float z  = 1.f / (1.f + __expf(-(gib[256 + j] + hz)));
      const float nn = tanhf(gib[512 + j] + r * hn);
      const float hv = (1.f - z) * nn + z * hS[b * H_STRIDE + j];
      hS[b * H_STRIDE + j] = hv;     // element-wise: same thread reads/writes
      yst[e] = hv;
    }
    __syncthreads();                 // protects hS / gS / giS for next step
  }
}

// ---------------------------------------------------------------------------
// Finalize: out1[t,b,0:256]=ysF1[t,b], out1[t,b,256:512]=ysB1 rolled back;
// zero-masked where t >= len[b].
// ---------------------------------------------------------------------------
__global__ __launch_bounds__(256) void finalize_out_kernel(
    const float* __restrict__ ysF1, const float* __restrict__ ysB1,
    const int* __restrict__ lengths, float* __restrict__ out) {
  const size_t idx = (size_t)blockIdx.x * 256 + threadIdx.x;  // T*B*512
  const int t = (int)(idx >> 14);           // / (32*512)
  const int r = (int)(idx & 16383);
  const int b = r >> 9;
  const int j = r & 511;
  const int len = lengths[b];
  float v = 0.f;
  if (t < len) {
    if (j < 256) {
      v = ysF1[((size_t)t * BB + b) * HH + j];
    } else {
      const int ts = (t + (TT - len)) & (TT - 1);
      v = ysB1[((size_t)ts * BB + b) * HH + (j - 256)];
    }
  }
  out[idx] = v;
}

__global__ __launch_bounds__(256) void finalize_hidden_kernel(
    const float* __restrict__ ysF0, const float* __restrict__ ysB0,
    const float* __restrict__ ysF1, const float* __restrict__ ysB1,
    const int* __restrict__ lengths, float* __restrict__ outH) {
  const int idx = blockIdx.x * 256 + threadIdx.x;   // 4*32*256
  const int d = idx >> 13;
  const int b = (idx >> 8) & 31;
  const int j = idx & 255;
  const int len = lengths[b];
  const float* src = (d == 0) ? ysF0 : (d == 1) ? ysB0 : (d == 2) ? ysF1 : ysB1;
  const int t = (d & 1) ? (TT - len) : (len - 1);
  outH[idx] = src[((size_t)t * BB + b) * HH + j];
}

// ---------------------------------------------------------------------------
extern "C" void kernel_launch(void* const* d_in, const int* in_sizes, int n_in,
                              void* d_out, int out_size, void* d_ws, size_t ws_size,
                              hipStream_t stream) {
  const float* x        = (const float*)d_in[0];
  const float* w_ih_l0f = (const float*)d_in[1];
  const float* w_hh_l0f = (const float*)d_in[2];
  const float* b_ih_l0f = (const float*)d_in[3];
  const float* b_hh_l0f = (const float*)d_in[4];
  const float* w_ih_l0b = (const float*)d_in[5];
  const float* w_hh_l0b = (const float*)d_in[6];
  const float* b_ih_l0b = (const float*)d_in[7];
  const float* b_hh_l0b = (const float*)d_in[8];
  const float* w_ih_l1f = (const float*)d_in[9];
  const float* w_hh_l1f = (const float*)d_in[10];
  const float* b_ih_l1f = (const float*)d_in[11];
  const float* b_hh_l1f = (const float*)d_in[12];
  const float* w_ih_l1b = (const float*)d_in[13];
  const float* w_hh_l1b = (const float*)d_in[14];
  const float* b_ih_l1b = (const float*)d_in[15];
  const float* b_hh_l1b = (const float*)d_in[16];
  const int*   lengths  = (const int*)d_in[17];

  float* ws   = (float*)d_ws;
  const size_t GI_SZ = (size_t)TT * BB * GG;   // 25,165,824 floats
  const size_t YS_SZ = (size_t)TT * BB * HH;   //  8,388,608 floats
  float* giF  = ws;
  float* giB  = giF + GI_SZ;
  float* ysF0 = giB + GI_SZ;
  float* ysB0 = ysF0 + YS_SZ;
  float* ysF1 = ysB0 + YS_SZ;
  float* ysB1 = ysF1 + YS_SZ;

  const dim3 giGrid(12288), giBlk(256);
  const size_t scanLds =
      (32 * H_STRIDE + 32 * G_STRIDE + 32 * GG) * sizeof(float); // 230,400 B

  // ---- layer 0 ----
  gi_gemm_kernel<<<giGrid, giBlk, 0, stream>>>(x, nullptr, 256, 0,  0, 0,
                                               w_ih_l0f, b_ih_l0f, lengths, giF);
  gi_gemm_kernel<<<giGrid, giBlk, 0, stream>>>(x, nullptr, 256, 0, -1, 0,
                                               w_ih_l0b, b_ih_l0b, lengths, giB);
  gru_scan_kernel<<<2, 1024, scanLds, stream>>>(giF, giB, w_hh_l0f, w_hh_l0b,
                                                b_hh_l0f, b_hh_l0b, ysF0, ysB0);
  // ---- layer 1 (backward half of out0 un-rolls under the l1b shift) ----
  gi_gemm_kernel<<<giGrid, giBlk, 0, stream>>>(ysF0, ysB0, 256, 256,  0, +1,
                                               w_ih_l1f, b_ih_l1f, lengths, giF);
  gi_gemm_kernel<<<giGrid, giBlk, 0, stream>>>(ysF0, ysB0, 256, 256, -1,  0,
                                               w_ih_l1b, b_ih_l1b, lengths, giB);
  gru_scan_kernel<<<2, 1024, scanLds, stream>>>(giF, giB, w_hh_l1f, w_hh_l1b,
                                                b_hh_l1f, b_hh_l1b, ysF1, ysB1);
  // ---- finalize ----
  finalize_out_kernel<<<65536, 256, 0, stream>>>(ysF1, ysB1, lengths, (float*)d_out);
  finalize_hidden_kernel<<<128, 256, 0, stream>>>(ysF0, ysB0, ysF1, ysB1, lengths,
                                                  (float*)d_out + (size_t)TT * BB * 512);
}